// RepMLPBlock_13993003450965
// MI455X (gfx1250) — compile-verified
//
#include <hip/hip_runtime.h>

#define EPS 1e-5f

typedef float v2f __attribute__((ext_vector_type(2)));
typedef float v8f __attribute__((ext_vector_type(8)));
typedef unsigned int tdmg0_t __attribute__((ext_vector_type(4)));
typedef int tdmg1_t __attribute__((ext_vector_type(8)));
typedef int tdmg2_t __attribute__((ext_vector_type(4)));

// ---------------- kernel 1: global average pool over H*W ----------------
__global__ __launch_bounds__(256) void pool_kernel(const float* __restrict__ x,
                                                   float* __restrict__ pooled) {
  __shared__ float red[256];
  const int t = threadIdx.x;
  const float* p = x + (size_t)blockIdx.x * 4096;
  float s = 0.f;
#pragma unroll
  for (int k = 0; k < 4; ++k) {
    float4 v = *(const float4*)(p + 4 * (t + 256 * k));
    s += v.x + v.y + v.z + v.w;
  }
  red[t] = s;
  __syncthreads();
  for (int off = 128; off > 0; off >>= 1) {
    if (t < off) red[t] += red[t + off];
    __syncthreads();
  }
  if (t == 0) pooled[blockIdx.x] = red[0] * (1.f / 4096.f);
}

// ---------------- kernel 2: gate MLP (tiny) ----------------
__global__ __launch_bounds__(256) void gate_kernel(const float* __restrict__ pooled,
                                                   const float* __restrict__ fc1w,
                                                   const float* __restrict__ fc1b,
                                                   const float* __restrict__ fc2w,
                                                   const float* __restrict__ fc2b,
                                                   float* __restrict__ gate) {
  __shared__ float pl[256];
  __shared__ float hbuf[64];
  const int b = blockIdx.x, t = threadIdx.x;
  pl[t] = pooled[b * 256 + t];
  __syncthreads();
  if (t < 64) {
    float acc = fc1b[t];
    for (int k = 0; k < 256; ++k) acc += pl[k] * fc1w[t * 256 + k];
    hbuf[t] = fmaxf(acc, 0.f);
  }
  __syncthreads();
  float acc = fc2b[t];
  for (int k = 0; k < 64; ++k) acc += hbuf[k] * fc2w[t * 64 + k];
  gate[b * 256 + t] = 1.f / (1.f + __expf(-acc));
}

// ---------------- kernel 3: fused partition-GEMM + reparam convs + gate ----------------
// One WG per (b, hp-band ip, shareset s). Loads the 64 channels c = c1*4+s,
// rows ip*8..ip*8+7, full width (contiguous 2KB per channel) via TDM.
// LDS x layout: per-channel 512 floats + 2-dword pad every 256 dwords (stride 516).
__global__ __launch_bounds__(256) void repmlp_main(
    const float* __restrict__ x, const float* __restrict__ fc3_w,
    const float* __restrict__ fc3g, const float* __restrict__ fc3b,
    const float* __restrict__ fc3m, const float* __restrict__ fc3v,
    const float* __restrict__ c1w, const float* __restrict__ bn1g,
    const float* __restrict__ bn1b, const float* __restrict__ bn1m,
    const float* __restrict__ bn1v, const float* __restrict__ c3w,
    const float* __restrict__ bn3g, const float* __restrict__ bn3b,
    const float* __restrict__ bn3m, const float* __restrict__ bn3v,
    const float* __restrict__ gate, float* __restrict__ out) {
  __shared__ __align__(16) float lds_x[64 * 516];   // 129 KB band (padded)
  __shared__ __align__(16) float lds_w[64 * 68];    // 17 KB weights, 68-float pitch
  __shared__ float lds_gate[8];

  const int bid = blockIdx.x;
  const int s = bid & 3;
  const int ip = (bid >> 2) & 7;
  const int b = bid >> 5;
  const int t = threadIdx.x;

#if __has_builtin(__builtin_amdgcn_tensor_load_to_lds)
  if (t < 32) {  // wave 0 issues one TDM descriptor for the whole 129KB tile
    unsigned lds_addr = (unsigned)(unsigned long long)(void*)&lds_x[0];
    unsigned long long ga = (unsigned long long)(uintptr_t)x +
                            4ull * ((unsigned long long)(b * 256 + s) * 4096ull +
                                    (unsigned long long)ip * 512ull);
    tdmg0_t g0;
    g0.x = 1u;                                        // count=1 (valid)
    g0.y = lds_addr;                                  // LDS byte address
    g0.z = (unsigned)(ga & 0xffffffffu);              // global addr lo
    g0.w = (unsigned)((ga >> 32) & 0x01ffffffu) | (2u << 30);  // hi + type=2
    tdmg1_t g1;
    g1[0] = (int)((2u << 16)     // data_size = 4B
                  | (1u << 20)   // pad_enable
                  | (7u << 22)   // pad_interval: every 256 dwords
                  | (1u << 25)); // pad_amount: 2 dwords  -> stride 516 floats
    g1[1] = (int)(512u << 16);   // tensor_dim0 = 512 (8 contiguous rows)
    g1[2] = (int)(64u << 16);    // tensor_dim1 = 64 channels
    g1[3] = (int)(512u << 16);   // tile_dim0 = 512
    g1[4] = 64;                  // tile_dim1 = 64, tile_dim2 = 0
    g1[5] = 16384;               // tensor_dim0_stride = 4 channels * 4096 elems
    g1[6] = 0;
    g1[7] = 0;
    tdmg2_t g2 = {0, 0, 0, 0};
    tdmg1_t g4 = {0, 0, 0, 0, 0, 0, 0, 0};
    __builtin_amdgcn_tensor_load_to_lds(g0, g1, g2, g2, g4, 0);
    __builtin_amdgcn_s_wait_tensorcnt(0);
  }
#else
  // fallback: cooperative coalesced loads (no TDM builtin on this toolchain)
  for (int k = 0; k < 32; ++k) {
    int fi = t + 256 * k;        // float4 index, 8192 total
    int c1 = fi >> 7;            // 128 float4 per channel
    int off = (fi & 127) * 4;
    float4 vv = *(const float4*)(x + (size_t)(b * 256 + c1 * 4 + s) * 4096 +
                                 ip * 512 + off);
    int d = c1 * 516 + off + 2 * (off >> 8);
    *(v2f*)(&lds_x[d]) = (v2f){vv.x, vv.y};
    *(v2f*)(&lds_x[d + 2]) = (v2f){vv.z, vv.w};
  }
#endif

  // stage fc3 weights for this shareset: lds_w[o][k], pitch 68
#pragma unroll
  for (int k = 0; k < 4; ++k) {
    int fi = t + 256 * k;        // 1024 float4 = 4096 floats
    int r = fi >> 4, c4 = fi & 15;
    float4 wv = *(const float4*)(fc3_w + (size_t)(s * 64 + r) * 64 + c4 * 4);
    *(float4*)(&lds_w[r * 68 + c4 * 4]) = wv;
  }
  if (t < 8) lds_gate[t] = gate[b * 256 + (ip * 8 + t) * 4 + s];

  // fold BN constants (uniform per WG)
  float sc3 = fc3g[s] * rsqrtf(fc3v[s] + EPS);
  float sh = fc3b[s] - fc3m[s] * sc3;
  float s1 = bn1g[s] * rsqrtf(bn1v[s] + EPS);
  float a1 = s1 * c1w[s];
  sh += bn1b[s] - bn1m[s] * s1;
  float sk = bn3g[s] * rsqrtf(bn3v[s] + EPS);
  sh += bn3b[s] - bn3m[s] * sk;
  float k3[9];
#pragma unroll
  for (int q = 0; q < 9; ++q) k3[q] = sk * c3w[s * 9 + q];

  __syncthreads();

  const int w = t >> 5, l = t & 31;
  const int lo = l & 15, hi = l >> 4;
  const v8f zero = {0.f, 0.f, 0.f, 0.f, 0.f, 0.f, 0.f, 0.f};

  for (int mi = 0; mi < 4; ++mi) {
    const int mt = w + 8 * mi;          // m-tile 0..31: rows r = mt*16 .. +15
    const int j = mt >> 2;              // wp partition index (uniform per tile)
    const int r0 = mt * 16;
    v8f acc[4];
#pragma unroll
    for (int nt = 0; nt < 4; ++nt) acc[nt] = zero;

    const int c1A = (r0 + lo) & 63;     // A rows: lanes 0-15 = M, both halves
    const float* lxA = &lds_x[c1A * 516];
#pragma unroll
    for (int kk = 0; kk < 16; ++kk) {
      const int k0 = 4 * kk + 2 * hi;   // this lane's K pair
      const int idx = (k0 >> 3) * 64 + j * 8 + (k0 & 7);
      v2f a = *(const v2f*)(lxA + idx + 2 * (idx >> 8));
#pragma unroll
      for (int nt = 0; nt < 4; ++nt) {
        v2f bf = *(const v2f*)(&lds_w[(nt * 16 + lo) * 68 + k0]);
        acc[nt] = __builtin_amdgcn_wmma_f32_16x16x4_f32(
            false, a, false, bf, (short)0, acc[nt], false, false);
      }
    }

    const float gj = lds_gate[j];
    float* ob = out + (size_t)(b * 256 + (ip * 8 + j) * 4 + s) * 4096;
#pragma unroll
    for (int nt = 0; nt < 4; ++nt) {
      const int pix = nt * 16 + lo;     // C/D layout: N = lane%16 (+16*nt)
      const int y = pix >> 3, xx = pix & 7;
#pragma unroll
      for (int v = 0; v < 8; ++v) {     // C/D layout: M = v + 8*(lane/16)
        const int c1 = (r0 + v + 8 * hi) & 63;
        const float* lx = &lds_x[c1 * 516];
        const int cidx = y * 64 + j * 8 + xx;
        float conv = a1 * lx[cidx + 2 * (cidx >> 8)];
#pragma unroll
        for (int dy = -1; dy <= 1; ++dy) {
          int yy = y + dy;
          if ((unsigned)yy < 8u) {
#pragma unroll
            for (int dx = -1; dx <= 1; ++dx) {
              int xn = xx + dx;
              if ((unsigned)xn < 8u) {
                int nidx = yy * 64 + j * 8 + xn;
                conv += k3[(dy + 1) * 3 + (dx + 1)] * lx[nidx + 2 * (nidx >> 8)];
              }
            }
          }
        }
        float val = (acc[nt][v] * sc3 + conv + sh) * gj;
        ob[((c1 >> 3) * 8 + y) * 64 + (c1 & 7) * 8 + xx] = val;
      }
    }
  }
}

extern "C" void kernel_launch(void* const* d_in, const int* in_sizes, int n_in,
                              void* d_out, int out_size, void* d_ws, size_t ws_size,
                              hipStream_t stream) {
  const float* x     = (const float*)d_in[0];
  const float* fc1_w = (const float*)d_in[1];
  const float* fc1_b = (const float*)d_in[2];
  const float* fc2_w = (const float*)d_in[3];
  const float* fc2_b = (const float*)d_in[4];
  const float* fc3_w = (const float*)d_in[5];
  const float* fc3g  = (const float*)d_in[6];
  const float* fc3b  = (const float*)d_in[7];
  const float* fc3m  = (const float*)d_in[8];
  const float* fc3v  = (const float*)d_in[9];
  const float* c1w   = (const float*)d_in[10];
  const float* bn1g  = (const float*)d_in[11];
  const float* bn1b  = (const float*)d_in[12];
  const float* bn1m  = (const float*)d_in[13];
  const float* bn1v  = (const float*)d_in[14];
  const float* c3w   = (const float*)d_in[15];
  const float* bn3g  = (const float*)d_in[16];
  const float* bn3b  = (const float*)d_in[17];
  const float* bn3m  = (const float*)d_in[18];
  const float* bn3v  = (const float*)d_in[19];

  float* pooled = (float*)d_ws;          // 32*256 floats
  float* gatebuf = pooled + 32 * 256;    // 32*256 floats

  pool_kernel<<<32 * 256, 256, 0, stream>>>(x, pooled);
  gate_kernel<<<32, 256, 0, stream>>>(pooled, fc1_w, fc1_b, fc2_w, fc2_b, gatebuf);
  repmlp_main<<<1024, 256, 0, stream>>>(x, fc3_w, fc3g, fc3b, fc3m, fc3v,
                                        c1w, bn1g, bn1b, bn1m, bn1v,
                                        c3w, bn3g, bn3b, bn3m, bn3v,
                                        gatebuf, (float*)d_out);
}